// RevGCN_20675972563256
// MI455X (gfx1250) — compile-verified
//
#include <hip/hip_runtime.h>
#include <hip/hip_bf16.h>
#include <math.h>

#define N_NODES 100000
#define N_EDGES 1600000
#define HCH 128
#define DCH 64
#define TT 112
#define MSG_EPS 1e-7f

typedef __attribute__((ext_vector_type(2))) float v2f;
typedef __attribute__((ext_vector_type(8))) float v8f;

// D = A(16x4,f32) * B(4x16,f32) + C(16x16,f32), full fp32 precision.
__device__ __forceinline__ v8f wmma4(v2f a, v2f b, v8f c) {
  return __builtin_amdgcn_wmma_f32_16x16x4_f32(false, a, false, b, (short)0, c,
                                               false, false);
}

// ---------------------------------------------------------------------------
// Node encoder: h[n] = [node_features[node_index[n]], x[n]@W_oh+b_oh] @ W_ne + b_ne
// One wave per 16-node tile. B staged TRANSPOSED in LDS so each WMMA B operand
// is one aligned ds_load_b64 (no register-shuffle movs).
// ---------------------------------------------------------------------------
__global__ __launch_bounds__(128) void encode_kernel(
    const float* __restrict__ x, const int* __restrict__ node_index,
    const float* __restrict__ node_features,
    const float* __restrict__ W_oh, const float* __restrict__ b_oh,
    const float* __restrict__ W_ne, const float* __restrict__ b_ne,
    float* __restrict__ h)
{
  __shared__ __align__(16) float Bt[128 * 16];   // Bt[col][k] = W_ne[k][col]
  __shared__ __align__(16) float As[4][16 * 16];
  const int tid = threadIdx.x;
  const int wid = tid >> 5, lane = tid & 31;
  const int lr = lane & 15, half = lane >> 4;

  #pragma unroll
  for (int k = 0; k < 16; ++k) Bt[tid * 16 + k] = W_ne[k * 128 + tid];

  const int tile = blockIdx.x * 4 + wid;
  const bool active = (tile * 16) < N_NODES;
  const int nb = tile * 16;

  if (active) {
    const int row = lr;
    if (half == 0) {               // cols 0..7: gathered node features
      const int ni = node_index[nb + row];
      #pragma unroll
      for (int k = 0; k < 8; ++k)
        As[wid][row * 16 + k] = node_features[(size_t)ni * 8 + k];
    } else {                       // cols 8..15: x @ W_oh + b_oh
      float xv[8];
      #pragma unroll
      for (int j = 0; j < 8; ++j) xv[j] = x[(size_t)(nb + row) * 8 + j];
      #pragma unroll
      for (int k = 0; k < 8; ++k) {
        float s = b_oh[k];
        #pragma unroll
        for (int j = 0; j < 8; ++j) s += xv[j] * W_oh[j * 8 + k];
        As[wid][row * 16 + 8 + k] = s;
      }
    }
  }
  __syncthreads();
  if (!active) return;

  v8f acc[8];
  #pragma unroll
  for (int ct = 0; ct < 8; ++ct) acc[ct] = (v8f){0.f,0.f,0.f,0.f,0.f,0.f,0.f,0.f};

  #pragma unroll
  for (int kk = 0; kk < 16; kk += 4) {
    const v2f a = *(const v2f*)&As[wid][lr * 16 + kk + 2 * half];
    #pragma unroll
    for (int ct = 0; ct < 8; ++ct) {
      const v2f b = *(const v2f*)&Bt[(ct * 16 + lr) * 16 + kk + 2 * half];
      acc[ct] = wmma4(a, b, acc[ct]);
    }
  }
  #pragma unroll
  for (int ct = 0; ct < 8; ++ct) {
    const int col = ct * 16 + lr;
    const float bias = b_ne[col];
    #pragma unroll
    for (int i = 0; i < 8; ++i)
      h[(size_t)(nb + i + 8 * half) * HCH + col] = acc[ct][i] + bias;
  }
}

// ---------------------------------------------------------------------------
// Fold edge encoder: Wc = W_ee(8x128) @ We[l,g](128x64);  bc = b_ee@We + be
// ---------------------------------------------------------------------------
__global__ __launch_bounds__(64) void prep_weights_kernel(
    const float* __restrict__ W_ee, const float* __restrict__ b_ee,
    const float* __restrict__ We, const float* __restrict__ be, int lg,
    float* __restrict__ Wc, float* __restrict__ bc)
{
  const int c = threadIdx.x;                 // 0..63
  const float* Wlg = We + (size_t)lg * HCH * DCH;
  float s[8] = {0.f, 0.f, 0.f, 0.f, 0.f, 0.f, 0.f, 0.f};
  float sb = 0.f;
  for (int k = 0; k < HCH; ++k) {
    const float w = Wlg[k * DCH + c];
    sb += b_ee[k] * w;
    #pragma unroll
    for (int r = 0; r < 8; ++r) s[r] += W_ee[r * HCH + k] * w;
  }
  #pragma unroll
  for (int r = 0; r < 8; ++r) Wc[r * DCH + c] = s[r];
  bc[c] = sb + be[lg * DCH + c];
}

// One-shot transpose of Wp(128x112) -> WpT(112x128) so the final projection's
// WMMA B operand is an aligned global_load_b64 from L2.
__global__ __launch_bounds__(256) void prep_wpt_kernel(
    const float* __restrict__ Wp, float* __restrict__ WpT)
{
  const int flat = blockIdx.x * 256 + threadIdx.x;
  if (flat >= TT * HCH) return;
  const int c = flat >> 7, k = flat & 127;   // WpT[c][k] = Wp[k][c]
  WpT[flat] = Wp[k * TT + c];
}

// ---------------------------------------------------------------------------
// Pre-activation: hb[n] = relu(layernorm(h[n, inOff:inOff+64], g, b))
// ---------------------------------------------------------------------------
__global__ __launch_bounds__(128) void prep_block_kernel(
    const float* __restrict__ h, int inOff,
    const float* __restrict__ g, const float* __restrict__ b,
    float* __restrict__ hb)
{
  const int wid = threadIdx.x >> 5, lane = threadIdx.x & 31;
  const int node = blockIdx.x * 4 + wid;
  if (node >= N_NODES) return;
  const float* row = h + (size_t)node * HCH + inOff;
  const float v0 = row[lane], v1 = row[lane + 32];
  float s = v0 + v1;
  #pragma unroll
  for (int m = 16; m >= 1; m >>= 1) s += __shfl_xor(s, m, 32);
  const float mu = s * (1.0f / 64.0f);
  const float d0 = v0 - mu, d1 = v1 - mu;
  float q = d0 * d0 + d1 * d1;
  #pragma unroll
  for (int m = 16; m >= 1; m >>= 1) q += __shfl_xor(q, m, 32);
  const float r = rsqrtf(q * (1.0f / 64.0f) + 1e-5f);
  hb[(size_t)node * DCH + lane]      = fmaxf(d0 * r * g[lane] + b[lane], 0.f);
  hb[(size_t)node * DCH + lane + 32] = fmaxf(d1 * r * g[lane + 32] + b[lane + 32], 0.f);
}

// ---------------------------------------------------------------------------
// Edge pass 1: msg = relu(hb[src] + ea@Wc + bc) + eps; atomic max into mx[dst].
// msg > 0 always => integer atomicMax on float bits is order-correct, and
// mx initialized to 0 reproduces the reference's isfinite->0 replacement.
// ---------------------------------------------------------------------------
__global__ __launch_bounds__(256) void edge_max_kernel(
    const int* __restrict__ ei, const float* __restrict__ ea,
    const float* __restrict__ hb, const float* __restrict__ Wc,
    const float* __restrict__ bc, float* __restrict__ mx)
{
  const int e = blockIdx.x * 8 + (threadIdx.x >> 5);
  if (e >= N_EDGES) return;
  const int lane = threadIdx.x & 31;
  const int s = ei[e], d = ei[N_EDGES + e];
  const float4 a0 = ((const float4*)ea)[(size_t)e * 2];      // global_load_b128
  const float4 a1 = ((const float4*)ea)[(size_t)e * 2 + 1];
  const float av[8] = {a0.x, a0.y, a0.z, a0.w, a1.x, a1.y, a1.z, a1.w};
  float m0 = bc[lane] + hb[(size_t)s * DCH + lane];
  float m1 = bc[lane + 32] + hb[(size_t)s * DCH + lane + 32];
  #pragma unroll
  for (int k = 0; k < 8; ++k) {
    m0 += av[k] * Wc[k * DCH + lane];
    m1 += av[k] * Wc[k * DCH + lane + 32];
  }
  m0 = fmaxf(m0, 0.f) + MSG_EPS;
  m1 = fmaxf(m1, 0.f) + MSG_EPS;
  atomicMax((int*)(mx + (size_t)d * DCH + lane), __float_as_int(m0));
  atomicMax((int*)(mx + (size_t)d * DCH + lane + 32), __float_as_int(m1));
}

// ---------------------------------------------------------------------------
// Edge pass 2: ex = exp(msg - mx[dst]); den += ex; num += msg*ex.
// (agg = num/(den+1e-16) folds softmax normalization out of the edge loop)
// ---------------------------------------------------------------------------
__global__ __launch_bounds__(256) void edge_sum_kernel(
    const int* __restrict__ ei, const float* __restrict__ ea,
    const float* __restrict__ hb, const float* __restrict__ Wc,
    const float* __restrict__ bc, const float* __restrict__ mx,
    float* __restrict__ den, float* __restrict__ num)
{
  const int e = blockIdx.x * 8 + (threadIdx.x >> 5);
  if (e >= N_EDGES) return;
  const int lane = threadIdx.x & 31;
  const int s = ei[e], d = ei[N_EDGES + e];
  const float4 a0 = ((const float4*)ea)[(size_t)e * 2];
  const float4 a1 = ((const float4*)ea)[(size_t)e * 2 + 1];
  const float av[8] = {a0.x, a0.y, a0.z, a0.w, a1.x, a1.y, a1.z, a1.w};
  float m0 = bc[lane] + hb[(size_t)s * DCH + lane];
  float m1 = bc[lane + 32] + hb[(size_t)s * DCH + lane + 32];
  #pragma unroll
  for (int k = 0; k < 8; ++k) {
    m0 += av[k] * Wc[k * DCH + lane];
    m1 += av[k] * Wc[k * DCH + lane + 32];
  }
  m0 = fmaxf(m0, 0.f) + MSG_EPS;
  m1 = fmaxf(m1, 0.f) + MSG_EPS;
  const float e0 = __expf(m0 - mx[(size_t)d * DCH + lane]);
  const float e1 = __expf(m1 - mx[(size_t)d * DCH + lane + 32]);
  atomicAdd(den + (size_t)d * DCH + lane, e0);
  atomicAdd(den + (size_t)d * DCH + lane + 32, e1);
  atomicAdd(num + (size_t)d * DCH + lane, m0 * e0);
  atomicAdd(num + (size_t)d * DCH + lane + 32, m1 * e1);
}

// ---------------------------------------------------------------------------
// Block output (WMMA): h[:,off:off+64] += (hb + num/(den+1e-16)) @ Wm + bm
// Wm staged TRANSPOSED in LDS -> B operand is one aligned ds_load_b64.
// ---------------------------------------------------------------------------
__global__ __launch_bounds__(128) void block_out_kernel(
    float* __restrict__ h, int off, const float* __restrict__ hb,
    const float* __restrict__ num, const float* __restrict__ den,
    const float* __restrict__ Wm, const float* __restrict__ bm)
{
  __shared__ __align__(16) float Bt[64 * 64];    // Bt[col][k] = Wm[k][col]
  __shared__ float bmS[64];
  __shared__ __align__(16) float As[4][16 * 64];
  const int tid = threadIdx.x, wid = tid >> 5, lane = tid & 31;
  const int lr = lane & 15, half = lane >> 4;

  #pragma unroll
  for (int t = 0; t < 32; ++t) {
    const int flat = t * 128 + tid;              // coalesced global read
    const int k = flat >> 6, c = flat & 63;
    Bt[c * 64 + k] = Wm[flat];
  }
  if (tid < 64) bmS[tid] = bm[tid];

  const int tile = blockIdx.x * 4 + wid;
  const bool active = (tile * 16) < N_NODES;
  const int nb = tile * 16;
  if (active) {
    #pragma unroll 4
    for (int j = 0; j < 32; ++j) {               // 16x64 tile, fully coalesced
      const int flat = j * 32 + lane;
      const size_t idx = (size_t)nb * DCH + flat;
      As[wid][flat] = hb[idx] + num[idx] / (den[idx] + 1e-16f);
    }
  }
  __syncthreads();
  if (!active) return;

  v8f acc[4];
  #pragma unroll
  for (int ct = 0; ct < 4; ++ct) acc[ct] = (v8f){0.f,0.f,0.f,0.f,0.f,0.f,0.f,0.f};

  for (int kk = 0; kk < 64; kk += 4) {
    const v2f a = *(const v2f*)&As[wid][lr * 64 + kk + 2 * half];
    #pragma unroll
    for (int ct = 0; ct < 4; ++ct) {
      const v2f b = *(const v2f*)&Bt[(ct * 16 + lr) * 64 + kk + 2 * half];
      acc[ct] = wmma4(a, b, acc[ct]);
    }
  }
  #pragma unroll
  for (int ct = 0; ct < 4; ++ct) {
    const int col = ct * 16 + lr;
    const float bias = bmS[col];
    #pragma unroll
    for (int i = 0; i < 8; ++i) {
      float* p = h + (size_t)(nb + i + 8 * half) * HCH + off + col;
      *p = *p + acc[ct][i] + bias;               // reversible residual in place
    }
  }
}

// ---------------------------------------------------------------------------
// Final: out = relu(layernorm(h, last_g, last_b)) @ WpT^T + bp
// Fused in-LDS LayerNorm (2 lanes/row, bank-rotated sweeps), then WMMA with
// B from pre-transposed WpT (aligned global_load_b64, L2-resident 57KB).
// ---------------------------------------------------------------------------
__global__ __launch_bounds__(128) void final_kernel(
    const float* __restrict__ h, const float* __restrict__ lgm,
    const float* __restrict__ lbt, const float* __restrict__ WpT,
    const float* __restrict__ bp, float* __restrict__ out)
{
  __shared__ __align__(16) float As[4][16 * 128];
  const int tid = threadIdx.x, wid = tid >> 5, lane = tid & 31;
  const int lr = lane & 15, half = lane >> 4;
  const int tile = blockIdx.x * 4 + wid;
  if ((tile * 16) >= N_NODES) return;            // per-wave exit; no block barriers
  const int nb = tile * 16;

  #pragma unroll 4
  for (int j = 0; j < 64; ++j) {                 // stage raw 16x128 tile, coalesced
    const int flat = j * 32 + lane;
    As[wid][flat] = h[(size_t)nb * HCH + flat];
  }
  asm volatile("s_wait_dscnt 0x0" ::: "memory"); // intra-wave LDS RAW fence

  const int row = lane >> 1, part = lane & 1;
  float s = 0.f;
  for (int t = 0; t < 64; ++t) {
    const int col = part * 64 + ((t + lane) & 63);
    s += As[wid][row * 128 + col];
  }
  s += __shfl_xor(s, 1, 32);
  const float mu = s * (1.0f / 128.0f);
  float q = 0.f;
  for (int t = 0; t < 64; ++t) {
    const int col = part * 64 + ((t + lane) & 63);
    const float dd = As[wid][row * 128 + col] - mu;
    q += dd * dd;
  }
  q += __shfl_xor(q, 1, 32);
  const float rr = rsqrtf(q * (1.0f / 128.0f) + 1e-5f);
  for (int t = 0; t < 64; ++t) {
    const int col = part * 64 + ((t + lane) & 63);
    const float v = As[wid][row * 128 + col];
    As[wid][row * 128 + col] = fmaxf((v - mu) * rr * lgm[col] + lbt[col], 0.f);
  }
  asm volatile("s_wait_dscnt 0x0" ::: "memory");

  v8f acc[7];
  #pragma unroll
  for (int ct = 0; ct < 7; ++ct) acc[ct] = (v8f){0.f,0.f,0.f,0.f,0.f,0.f,0.f,0.f};

  for (int kk = 0; kk < 128; kk += 4) {
    const v2f a = *(const v2f*)&As[wid][lr * 128 + kk + 2 * half];
    #pragma unroll
    for (int ct = 0; ct < 7; ++ct) {
      const v2f b = *(const v2f*)&WpT[(size_t)(ct * 16 + lr) * 128 + kk + 2 * half];
      acc[ct] = wmma4(a, b, acc[ct]);
    }
  }
  #pragma unroll
  for (int ct = 0; ct < 7; ++ct) {
    const int col = ct * 16 + lr;
    const float bias = bp[col];
    #pragma unroll
    for (int i = 0; i < 8; ++i)
      out[(size_t)(nb + i + 8 * half) * TT + col] = acc[ct][i] + bias;
  }
}

// ---------------------------------------------------------------------------
extern "C" void kernel_launch(void* const* d_in, const int* in_sizes, int n_in,
                              void* d_out, int out_size, void* d_ws, size_t ws_size,
                              hipStream_t stream)
{
  (void)in_sizes; (void)n_in; (void)out_size; (void)ws_size;
  const float* x      = (const float*)d_in[0];
  const int*   nidx   = (const int*)d_in[1];
  const int*   eidx   = (const int*)d_in[2];
  const float* eattr  = (const float*)d_in[3];
  const float* nfeat  = (const float*)d_in[4];
  const float* W_oh   = (const float*)d_in[5];
  const float* b_oh   = (const float*)d_in[6];
  const float* W_ne   = (const float*)d_in[7];
  const float* b_ne   = (const float*)d_in[8];
  const float* W_ee   = (const float*)d_in[9];
  const float* b_ee   = (const float*)d_in[10];
  const float* ln_g   = (const float*)d_in[11];
  const float* ln_b   = (const float*)d_in[12];
  const float* We     = (const float*)d_in[13];
  const float* be     = (const float*)d_in[14];
  const float* Wm     = (const float*)d_in[15];
  const float* bm     = (const float*)d_in[16];
  const float* last_g = (const float*)d_in[17];
  const float* last_b = (const float*)d_in[18];
  const float* Wp     = (const float*)d_in[19];
  const float* bp     = (const float*)d_in[20];
  float* out = (float*)d_out;

  // Workspace: ~154 MB of fp32 scratch
  float* h   = (float*)d_ws;                       // [N,128]
  float* hb  = h   + (size_t)N_NODES * HCH;        // [N,64]
  float* mx  = hb  + (size_t)N_NODES * DCH;        // [N,64]
  float* den = mx  + (size_t)N_NODES * DCH;        // [N,64]
  float* num = den + (size_t)N_NODES * DCH;        // [N,64]
  float* Wc  = num + (size_t)N_NODES * DCH;        // [8,64]
  float* bc  = Wc + 8 * DCH;                       // [64]
  float* WpT = bc + DCH;                           // [112,128]

  const int nodeTiles  = (N_NODES + 15) / 16;      // 6250
  const int tileBlocks = (nodeTiles + 3) / 4;      // 1563
  const int edgeBlocks = (N_EDGES + 7) / 8;        // 200000
  const int rowBlocks  = (N_NODES + 3) / 4;        // 25000

  encode_kernel<<<tileBlocks, 128, 0, stream>>>(x, nidx, nfeat, W_oh, b_oh,
                                                W_ne, b_ne, h);
  prep_wpt_kernel<<<(TT * HCH + 255) / 256, 256, 0, stream>>>(Wp, WpT);

  for (int l = 0; l < 2; ++l) {
    for (int g = 0; g < 2; ++g) {
      const int lg = l * 2 + g;
      const int inOff  = (g == 0) ? 64 : 0;   // F0 reads x2, F1 reads y1
      const int outOff = (g == 0) ? 0 : 64;   // y1 -> left half, y2 -> right half
      prep_weights_kernel<<<1, 64, 0, stream>>>(W_ee, b_ee, We, be, lg, Wc, bc);
      prep_block_kernel<<<rowBlocks, 128, 0, stream>>>(h, inOff,
          ln_g + (size_t)lg * DCH, ln_b + (size_t)lg * DCH, hb);
      hipMemsetAsync(mx, 0, (size_t)3 * N_NODES * DCH * sizeof(float), stream);
      edge_max_kernel<<<edgeBlocks, 256, 0, stream>>>(eidx, eattr, hb, Wc, bc, mx);
      edge_sum_kernel<<<edgeBlocks, 256, 0, stream>>>(eidx, eattr, hb, Wc, bc,
                                                      mx, den, num);
      block_out_kernel<<<tileBlocks, 128, 0, stream>>>(h, outOff, hb, num, den,
          Wm + (size_t)lg * DCH * DCH, bm + (size_t)lg * DCH);
    }
  }
  final_kernel<<<tileBlocks, 128, 0, stream>>>(h, last_g, last_b, WpT, bp, out);
}